// SageEncoder_27788438405844
// MI455X (gfx1250) — compile-verified
//
#include <hip/hip_runtime.h>
#include <hip/hip_bf16.h>

typedef __attribute__((ext_vector_type(2))) float v2f;
typedef __attribute__((ext_vector_type(4))) float v4f;
typedef __attribute__((ext_vector_type(8))) float v8f;

#define DIM   256
#define HALF  128   // DIM/2, output half-width per weight matrix
#define DIM4  64    // DIM / 4 (float4 columns per row)

// ---------------------------------------------------------------------------
// Mean of `fan` gathered feature rows:  out[row, :] = mean_j feat[idx[row*fan+j], :]
// 256-thread block = 4 rows x 64 lanes; each lane moves one float4 (b128)
// per gathered row -> 512B per wave-instruction, minimizing issue pressure on
// the L2-bandwidth-bound gather (feature table is L2-resident: 102MB < 192MB).
// ---------------------------------------------------------------------------
__global__ __launch_bounds__(256) void mean_gather_kernel(
    const float* __restrict__ feat, const int* __restrict__ idx,
    float* __restrict__ out, int fan, float inv)
{
    const int t   = threadIdx.x;
    const int row = blockIdx.x * 4 + (t >> 6);
    const int q   = t & 63;                       // float4 column
    const v4f* f4 = (const v4f*)feat;
    const int* ip = idx + (size_t)row * fan;
    v4f acc = {};
    #pragma unroll 5
    for (int j = 0; j < fan; ++j) {
        acc += f4[(size_t)ip[j] * DIM4 + q];
    }
    ((v4f*)out)[(size_t)row * DIM4 + q] = acc * inv;
}

// Mean of `fan` contiguous rows (no gather), float4-wide.
__global__ __launch_bounds__(256) void mean_rows_kernel(
    const float* __restrict__ src, float* __restrict__ out, int fan, float inv)
{
    const int t   = threadIdx.x;
    const int row = blockIdx.x * 4 + (t >> 6);
    const int q   = t & 63;
    const v4f* s4 = (const v4f*)src;
    v4f acc = {};
    #pragma unroll
    for (int j = 0; j < fan; ++j) {
        acc += s4[((size_t)row * fan + j) * DIM4 + q];
    }
    ((v4f*)out)[(size_t)row * DIM4 + q] = acc * inv;
}

// ---------------------------------------------------------------------------
// One SAGE layer, fused:
//   out[m, 0:128]   = act( selfRow(m) @ Wself )
//   out[m, 128:256] = act( aggB[m]    @ Wneigh )
// selfRow(m) = srcA[idx[m]] (gather) or srcA[m] (idx == nullptr).
//
// Block = 512 threads = 16 wave32s, handles 16 rows of M.
//   - stages 16x256 self tile + 16x256 neigh tile in LDS (32 KB), b128-wide
//   - waves 0..7  : self half (Wself), N-tile = wave
//   - waves 8..15 : neigh half (Wneigh), N-tile = wave-8
// Each wave: 64 x v_wmma_f32_16x16x4_f32 over K=256 into a v8f accumulator.
// M is always a multiple of 16 -> EXEC all-1s for every WMMA (ISA req).
//
// A-operand layout (32-bit A 16x4): lane m=lane%16; VGPR0 holds K=0|K=2 by
// lane-half, VGPR1 holds K=1|K=3 -> per step lane loads A[m][k+2*half .. +1]
// as one 8B-aligned ds_load_b64. B mirrored (lane n=lane%16). C/D: VGPR r
// holds D[r + 8*half][n].
// ---------------------------------------------------------------------------
__global__ __launch_bounds__(512) void sage_layer_kernel(
    const float* __restrict__ srcA,    // self source rows (features / n0)
    const int*  __restrict__ idx,      // gather indices or nullptr
    const float* __restrict__ aggB,    // [M, 256] pre-aggregated neighbors
    const float* __restrict__ Wself,   // [256, 128] row-major
    const float* __restrict__ Wneigh,  // [256, 128] row-major
    float* __restrict__ out,           // [M, 256]
    int use_relu)
{
    __shared__ __align__(16) float ldsA[16 * DIM];   // self tile
    __shared__ __align__(16) float ldsN[16 * DIM];   // neigh tile

    const int tid       = threadIdx.x;
    const int block_row = blockIdx.x * 16;

    // Cooperative stage, b128-wide: 16*64 float4 per buffer, 512 threads.
    for (int i = tid; i < 16 * DIM4; i += 512) {
        const int r = i >> 6;          // row 0..15
        const int c = i & (DIM4 - 1);  // float4 col
        const int m = block_row + r;
        const int srow = idx ? idx[m] : m;
        ((v4f*)ldsA)[i] = ((const v4f*)srcA)[(size_t)srow * DIM4 + c];
        ((v4f*)ldsN)[i] = ((const v4f*)aggB)[(size_t)m   * DIM4 + c];
    }
    __syncthreads();

    const int wave = tid >> 5;         // 0..15
    const int lane = tid & 31;
    const int half = lane >> 4;        // 0/1 (lane-half)
    const int lidx = lane & 15;        // m for A, n for B/C/D

    const bool  self_half = (wave < 8);
    const float* W  = self_half ? Wself : Wneigh;
    const float* At = self_half ? (const float*)ldsA : (const float*)ldsN;
    const int ntile = wave & 7;
    const int bcol  = ntile * 16 + lidx;      // column within the 128-wide half

    v8f acc = {};
    #pragma unroll 8
    for (int k = 0; k < DIM; k += 4) {
        const int ka = k + half * 2;
        // A: lane (m=lidx) supplies A[m][ka], A[m][ka+1]  (8B aligned)
        v2f a = *(const v2f*)&At[lidx * DIM + ka];
        // B: lane (n=bcol) supplies W[ka][n], W[ka+1][n]
        v2f b;
        b.x = W[(size_t)ka       * HALF + bcol];
        b.y = W[(size_t)(ka + 1) * HALF + bcol];
        acc = __builtin_amdgcn_wmma_f32_16x16x4_f32(
            /*neg_a=*/false, a, /*neg_b=*/false, b,
            /*c_mod=*/(short)0, acc, /*reuse_a=*/false, /*reuse_b=*/false);
    }

    const int out_col = (self_half ? 0 : HALF) + bcol;
    #pragma unroll
    for (int r = 0; r < 8; ++r) {
        float v = acc[r];
        if (use_relu) v = fmaxf(v, 0.0f);
        const int m = block_row + r + 8 * half;
        out[(size_t)m * DIM + out_col] = v;
    }
}

// ---------------------------------------------------------------------------
// Launch: GraphSAGE 2-layer forward.
//   inputs: features, sn0, sn1, sn2, W0_self, W0_neigh, W1_self, W1_neigh
// ---------------------------------------------------------------------------
extern "C" void kernel_launch(void* const* d_in, const int* in_sizes, int n_in,
                              void* d_out, int out_size, void* d_ws, size_t ws_size,
                              hipStream_t stream) {
    const float* features = (const float*)d_in[0];
    const int*   sn0      = (const int*)d_in[1];
    const int*   sn1      = (const int*)d_in[2];
    const int*   sn2      = (const int*)d_in[3];
    const float* W0s      = (const float*)d_in[4];
    const float* W0n      = (const float*)d_in[5];
    const float* W1s      = (const float*)d_in[6];
    const float* W1n      = (const float*)d_in[7];
    float* out = (float*)d_out;

    const int B  = in_sizes[1];            // 1024
    const int M1 = in_sizes[2];            // B * f0 = 10240
    const int f0 = M1 / B;                 // 10
    const int f1 = in_sizes[3] / M1;       // 25

    // Workspace carve-up (floats), ~24 MB total
    float* agg2   = (float*)d_ws;                    // [M1, 256]
    float* n1     = agg2   + (size_t)M1 * DIM;       // [M1, 256]
    float* agg1   = n1     + (size_t)M1 * DIM;       // [B, 256]
    float* n0     = agg1   + (size_t)B  * DIM;       // [B, 256]
    float* agg_n1 = n0     + (size_t)B  * DIM;       // [B, 256]

    // 1) agg2 = mean over f1 of features[sn2]          [M1, 256]
    mean_gather_kernel<<<M1 / 4, 256, 0, stream>>>(features, sn2, agg2, f1, 1.0f / f1);
    // 2) agg1 = mean over f0 of features[sn1]          [B, 256]
    mean_gather_kernel<<<B / 4, 256, 0, stream>>>(features, sn1, agg1, f0, 1.0f / f0);
    // 3) n1 = [relu(h1@W0s), relu(agg2@W0n)]           [M1, 256]
    sage_layer_kernel<<<M1 / 16, 512, 0, stream>>>(features, sn1, agg2, W0s, W0n, n1, 1);
    // 4) n0 = [relu(h0@W0s), relu(agg1@W0n)]           [B, 256]
    sage_layer_kernel<<<B / 16, 512, 0, stream>>>(features, sn0, agg1, W0s, W0n, n0, 1);
    // 5) agg_n1 = mean over f0 of n1                   [B, 256]
    mean_rows_kernel<<<B / 4, 256, 0, stream>>>(n1, agg_n1, f0, 1.0f / f0);
    // 6) out = [n0@W1s, agg_n1@W1n]                    [B, 256]
    sage_layer_kernel<<<B / 16, 512, 0, stream>>>(n0, nullptr, agg_n1, W1s, W1n, out, 0);
}